// DiffPool_62964220560012
// MI455X (gfx1250) — compile-verified
//
#include <hip/hip_runtime.h>
#include <math.h>

typedef __attribute__((ext_vector_type(2))) float v2f;
typedef __attribute__((ext_vector_type(8))) float v8f;

#define BM 128
#define BN 128
#define BK 16
#define ST_KM 132    // [k][m] LDS stride (transA A-tiles and B-tiles), floats
#define ST_MK 20     // [m][k] LDS stride (non-trans A-tiles), floats
#define A_BUF_F 2560 // floats per A buffer: max(16*132=2112, 128*20=2560)
#define B_BUF_F 2112 // floats per B buffer: 16*132

// flags: bit1 = self-loop epilogue: += (1 - diag[m]) * B[m][n]  (requires M == K)
//        bit2 = add bias[n]
//        bit3 = ReLU
#define GF_TRANSA 1
#define GF_SELF   2
#define GF_BIAS   4
#define GF_RELU   8

template <bool TRANSA>
__global__ __launch_bounds__(256) void wmma_gemm(
    const float* __restrict__ A, const float* __restrict__ B, float* __restrict__ C,
    int M, int N, int K, int lda, int ldb, int ldc,
    int flags, const float* __restrict__ bias, const float* __restrict__ diag)
{
    // double-buffered tiles, filled by async global->LDS DMA (ASYNCcnt)
    __shared__ __align__(16) float AsRaw[2][A_BUF_F];
    __shared__ __align__(16) float BsRaw[2][B_BUF_F];

    const int tid  = threadIdx.x;
    const int lane = tid & 31;
    const int wid  = tid >> 5;
    const int h    = lane >> 4;        // half-wave: 0 or 1
    const int ln   = lane & 15;
    const int wRow = (wid >> 2) * 64;  // wave row offset within block tile (0,64)
    const int wCol = (wid & 3) * 32;   // wave col offset (0,32,64,96)
    const int row0 = blockIdx.y * BM;
    const int col0 = blockIdx.x * BN;

    // low 32 bits of a generic LDS pointer are the raw LDS byte address
    const unsigned asBase = (unsigned)(uintptr_t)&AsRaw[0][0];
    const unsigned bsBase = (unsigned)(uintptr_t)&BsRaw[0][0];

    v8f acc[4][2];
    const v8f vzero = {0.f,0.f,0.f,0.f,0.f,0.f,0.f,0.f};
#pragma unroll
    for (int mi = 0; mi < 4; ++mi)
#pragma unroll
        for (int ni = 0; ni < 2; ++ni) acc[mi][ni] = vzero;

    // ---- async DMA of one K-block tile into LDS buffer `buf` ----
    // each thread issues 2 x b128 for A and 2 x b128 for B (512 float4 chunks each)
    auto issue_tile = [&](int kb, int buf) {
#pragma unroll
        for (int j = 0; j < 2; ++j) {
            const int c = tid + 256 * j;
            unsigned aoff;
            const float* ga;
            if (TRANSA) {            // A[K x M]: chunk = (k, 4 consecutive m) -> [k][m]
                int k = c >> 5, m4 = (c & 31) * 4;
                aoff = asBase + (unsigned)(buf * A_BUF_F + k * ST_KM + m4) * 4u;
                ga = &A[(size_t)(kb + k) * lda + (row0 + m4)];
            } else {                 // A[M x K]: chunk = (m, 4 consecutive k) -> [m][k]
                int m = c >> 2, kc = (c & 3) * 4;
                aoff = asBase + (unsigned)(buf * A_BUF_F + m * ST_MK + kc) * 4u;
                ga = &A[(size_t)(row0 + m) * lda + (kb + kc)];
            }
            asm volatile("global_load_async_to_lds_b128 %0, %1, off"
                         :: "v"(aoff), "v"((unsigned long long)(uintptr_t)ga) : "memory");

            const int k = c >> 5, n4 = (c & 31) * 4;   // B chunk = (k, 4 consecutive n)
            unsigned boff = bsBase + (unsigned)(buf * B_BUF_F + k * ST_KM + n4) * 4u;
            const float* gb = &B[(size_t)(kb + k) * ldb + (col0 + n4)];
            asm volatile("global_load_async_to_lds_b128 %0, %1, off"
                         :: "v"(boff), "v"((unsigned long long)(uintptr_t)gb) : "memory");
        }
    };

    // ---- software pipeline ----
    issue_tile(0, 0);

    const int nkb = K / BK;
    for (int ib = 0; ib < nkb; ++ib) {
        const int cur = ib & 1;
        asm volatile("s_wait_asynccnt 0" ::: "memory");  // my share of tile `cur` landed
        __syncthreads();  // all shares landed; all reads of buffer cur^1 are finished
        if (ib + 1 < nkb) issue_tile((ib + 1) * BK, cur ^ 1);  // DMA next tile during compute

        const float* __restrict__ Acur = &AsRaw[cur][0];
        const float* __restrict__ Bcur = &BsRaw[cur][0];
#pragma unroll
        for (int ks = 0; ks < 4; ++ks) {
            const int kk = ks * 4 + 2 * h;  // lanes 0-15: K={k0,k0+1}; lanes 16-31: K={k0+2,k0+3}
            v2f a[4], b[2];
#pragma unroll
            for (int mi = 0; mi < 4; ++mi) {
                int m = wRow + mi * 16 + ln;
                if (TRANSA) {
                    a[mi].x = Acur[kk * ST_KM + m];
                    a[mi].y = Acur[(kk + 1) * ST_KM + m];
                } else {        // contiguous 8B pair -> ds_load_b64, conflict-free (stride 20)
                    a[mi] = *(const v2f*)&Acur[m * ST_MK + kk];
                }
            }
#pragma unroll
            for (int ni = 0; ni < 2; ++ni) {
                int n = wCol + ni * 16 + ln;
                b[ni].x = Bcur[kk * ST_KM + n];
                b[ni].y = Bcur[(kk + 1) * ST_KM + n];
            }
#pragma unroll
            for (int mi = 0; mi < 4; ++mi)
#pragma unroll
                for (int ni = 0; ni < 2; ++ni)
                    acc[mi][ni] = __builtin_amdgcn_wmma_f32_16x16x4_f32(
                        false, a[mi], false, b[ni], (short)0, acc[mi][ni], false, false);
        }
    }

    const bool doSelf = (flags & GF_SELF) != 0;
    const bool doBias = (flags & GF_BIAS) != 0;
    const bool doRelu = (flags & GF_RELU) != 0;
#pragma unroll
    for (int mi = 0; mi < 4; ++mi)
#pragma unroll
        for (int ni = 0; ni < 2; ++ni)
#pragma unroll
            for (int r = 0; r < 8; ++r) {
                int gm = row0 + wRow + mi * 16 + r + 8 * h;  // C layout: vgpr r -> M = r (+8 hi half)
                int gn = col0 + wCol + ni * 16 + ln;
                float v = acc[mi][ni][r];
                if (doSelf) v += (1.0f - diag[gm]) * B[(size_t)gm * ldb + gn];
                if (doBias) v += bias[gn];
                if (doRelu) v = fmaxf(v, 0.0f);
                C[(size_t)gm * ldc + gn] = v;
            }
}

// BatchNorm (training-mode, biased var) over rows; one thread per column, deterministic.
__global__ void bn_kernel(float* __restrict__ Hm, int M, int N,
                          const float* __restrict__ gamma, const float* __restrict__ beta)
{
    int col = blockIdx.x * blockDim.x + threadIdx.x;
    if (col >= N) return;
    float s = 0.f, ss = 0.f;
    for (int r = 0; r < M; ++r) {
        float v = Hm[(size_t)r * N + col];
        s += v; ss += v * v;
    }
    float mu  = s / (float)M;
    float var = ss / (float)M - mu * mu;
    float sc  = gamma[col] * rsqrtf(var + 1e-5f);
    float sh  = beta[col] - mu * sc;
    for (int r = 0; r < M; ++r) {
        size_t idx = (size_t)r * N + col;
        Hm[idx] = Hm[idx] * sc + sh;
    }
}

// Row softmax in place + per-row entropy  -sum p log(p+1e-15)
__global__ __launch_bounds__(256) void softmax_ent(float* __restrict__ S, int C,
                                                   float* __restrict__ entrow)
{
    __shared__ float red[256];
    const int row = blockIdx.x, tid = threadIdx.x;
    float* Srow = S + (size_t)row * C;

    float m = -3.402823466e38f;
    for (int j = tid; j < C; j += 256) m = fmaxf(m, Srow[j]);
    red[tid] = m; __syncthreads();
    for (int s = 128; s > 0; s >>= 1) { if (tid < s) red[tid] = fmaxf(red[tid], red[tid + s]); __syncthreads(); }
    m = red[0]; __syncthreads();

    float sum = 0.f;
    for (int j = tid; j < C; j += 256) { float e = expf(Srow[j] - m); Srow[j] = e; sum += e; }
    red[tid] = sum; __syncthreads();
    for (int s = 128; s > 0; s >>= 1) { if (tid < s) red[tid] += red[tid + s]; __syncthreads(); }
    float inv = 1.0f / red[0]; __syncthreads();

    float ent = 0.f;
    for (int j = tid; j < C; j += 256) {
        float p = Srow[j] * inv; Srow[j] = p;
        ent -= p * logf(p + 1e-15f);
    }
    red[tid] = ent; __syncthreads();
    for (int s = 128; s > 0; s >>= 1) { if (tid < s) red[tid] += red[tid + s]; __syncthreads(); }
    if (tid == 0) entrow[row] = red[0];
}

__global__ __launch_bounds__(256) void reduce_partial(const float* __restrict__ src, size_t n,
                                                      int sq, float* __restrict__ partial)
{
    __shared__ float red[256];
    size_t i = (size_t)blockIdx.x * 256 + threadIdx.x;
    size_t stride = (size_t)gridDim.x * 256;
    float s = 0.f;
    for (; i < n; i += stride) { float v = src[i]; s += sq ? v * v : v; }
    red[threadIdx.x] = s; __syncthreads();
    for (int t = 128; t > 0; t >>= 1) { if ((int)threadIdx.x < t) red[threadIdx.x] += red[threadIdx.x + t]; __syncthreads(); }
    if (threadIdx.x == 0) partial[blockIdx.x] = red[0];
}

__global__ __launch_bounds__(256) void reduce_final(const float* __restrict__ partial, int n,
                                                    float scale, float* __restrict__ out)
{
    __shared__ float red[256];
    float s = 0.f;
    for (int i = threadIdx.x; i < n; i += 256) s += partial[i];
    red[threadIdx.x] = s; __syncthreads();
    for (int t = 128; t > 0; t >>= 1) { if ((int)threadIdx.x < t) red[threadIdx.x] += red[threadIdx.x + t]; __syncthreads(); }
    if (threadIdx.x == 0) *out = red[0] * scale;
}

__global__ __launch_bounds__(256) void trace_kernel(const float* __restrict__ A, int n,
                                                    float* __restrict__ out)
{
    __shared__ float red[256];
    float s = 0.f;
    for (int i = threadIdx.x; i < n; i += 256) s += A[(size_t)i * n + i];
    red[threadIdx.x] = s; __syncthreads();
    for (int t = 128; t > 0; t >>= 1) { if ((int)threadIdx.x < t) red[threadIdx.x] += red[threadIdx.x + t]; __syncthreads(); }
    if (threadIdx.x == 0) *out = red[0];
}

__global__ void extract_diag(const float* __restrict__ A, int n, float* __restrict__ d)
{
    int i = blockIdx.x * 256 + threadIdx.x;
    if (i < n) d[i] = A[(size_t)i * n + i];
}

// readout mean -> lin1+ReLU -> lin2 ; writes logits to out[0..1]
__global__ __launch_bounds__(512) void head_kernel(const float* __restrict__ H3,
    const float* __restrict__ W1, const float* __restrict__ b1,
    const float* __restrict__ W2, const float* __restrict__ b2,
    float* __restrict__ out)
{
    __shared__ float g[512];
    __shared__ float y[512];
    const int t = threadIdx.x;
    float s = 0.f;
    for (int r = 0; r < 512; ++r) s += H3[r * 512 + t];
    g[t] = s * (1.0f / 512.0f);
    __syncthreads();
    float a = b1[t];
    for (int k = 0; k < 512; ++k) a += g[k] * W1[k * 512 + t];
    y[t] = fmaxf(a, 0.0f);
    __syncthreads();
    if (t < 2) {
        float o = b2[t];
        for (int k = 0; k < 512; ++k) o += y[k] * W2[k * 2 + t];
        out[t] = o;
    }
}

// link = sqrt(max(||adj||^2 - 2 tr(adj_p) + ||s^T s||^2, 0)) / adj.size
__global__ void finish_kernel(const float* __restrict__ scal, float* __restrict__ out)
{
    float l1 = sqrtf(fmaxf(scal[0] - 2.0f * scal[1] + scal[2], 0.0f)) * (1.0f / (8192.0f * 8192.0f));
    float l2 = sqrtf(fmaxf(scal[3] - 2.0f * scal[4] + scal[5], 0.0f)) * (1.0f / (2048.0f * 2048.0f));
    out[2] = l1 + l2;
    out[3] = scal[6] + scal[7];
}

// ---------------------------------------------------------------------------

static inline void launch_gemm(hipStream_t st, const float* A, const float* B, float* C,
                               int M, int N, int K, int lda, int ldb, int ldc,
                               int flags, const float* bias, const float* diag)
{
    dim3 grid(N / BN, M / BM), block(256);
    if (flags & GF_TRANSA)
        wmma_gemm<true><<<grid, block, 0, st>>>(A, B, C, M, N, K, lda, ldb, ldc, flags, bias, diag);
    else
        wmma_gemm<false><<<grid, block, 0, st>>>(A, B, C, M, N, K, lda, ldb, ldc, flags, bias, diag);
}

// 3-layer GCN block: for each layer  h = BN(relu(adj_sl @ (x W) + b))
static void run_gnn(hipStream_t st, void* const* d_in, int pbase,
                    const float* X, int nodes, int fin, const int* cout3,
                    const float* ADJ, const float* DIAG,
                    float* Tbuf, float* Hbuf, float* Hfinal)
{
    const float* cur = X;
    int cin = fin;
    for (int l = 0; l < 3; ++l) {
        int co = cout3[l];
        const float* Wl = (const float*)d_in[pbase + l];
        const float* bl = (const float*)d_in[pbase + 3 + l];
        const float* gl = (const float*)d_in[pbase + 6 + l];
        const float* el = (const float*)d_in[pbase + 9 + l];
        float* Hl = (l == 2) ? Hfinal : Hbuf;
        // t = cur @ W
        launch_gemm(st, cur, Wl, Tbuf, nodes, co, cin, cin, co, co, 0, nullptr, nullptr);
        // h = relu(adj @ t + (1-diag)*t + b)   (self-loops folded into epilogue)
        launch_gemm(st, ADJ, Tbuf, Hl, nodes, co, nodes, nodes, co, co,
                    GF_SELF | GF_BIAS | GF_RELU, bl, DIAG);
        bn_kernel<<<dim3(co / 256), dim3(256), 0, st>>>(Hl, nodes, co, gl, el);
        cur = Hl;
        cin = co;
    }
}

extern "C" void kernel_launch(void* const* d_in, const int* in_sizes, int n_in,
                              void* d_out, int out_size, void* d_ws, size_t ws_size,
                              hipStream_t stream)
{
    const int N1 = 8192, FIN = 1024, H = 512, C1 = 2048, C2 = 512;
    const float* x   = (const float*)d_in[0];
    const float* adj = (const float*)d_in[1];
    // params flatten order (setup_inputs insertion order):
    // per gnn: W0,W1,W2, b0,b1,b2, g0,g1,g2, be0,be1,be2
    const int g1p = 2, g1e = 14, g2p = 26, g2e = 38, g3e = 50;
    const float* lin1_w = (const float*)d_in[62];
    const float* lin1_b = (const float*)d_in[63];
    const float* lin2_w = (const float*)d_in[64];
    const float* lin2_b = (const float*)d_in[65];
    float* out = (float*)d_out;

    // ---- workspace layout (floats) ----
    float* W = (float*)d_ws;
    size_t off = 0;
    auto alloc = [&](size_t n) { float* p = W + off; off += n; return p; };
    float* B1    = alloc((size_t)N1 * C1);   // big temp (t buffers, adj@s)
    float* B2    = alloc((size_t)N1 * H);    // h buffers / h_embed
    float* B3    = alloc((size_t)N1 * C1);   // s logits -> s
    float* ADJ1  = alloc((size_t)C1 * C1);
    float* G1    = alloc((size_t)C1 * C1);
    float* XP    = alloc((size_t)C1 * H);
    float* T2    = alloc((size_t)C1 * H);
    float* H2    = alloc((size_t)C1 * H);
    float* S2    = alloc((size_t)C1 * H);
    float* XP2   = alloc((size_t)C2 * H);
    float* ADJ2  = alloc((size_t)C2 * C2);
    float* G2    = alloc((size_t)C2 * C2);
    float* DIAGA = alloc(N1);
    float* DIAG1 = alloc(C1);
    float* DIAG2 = alloc(C2);
    float* ENT   = alloc(N1);
    float* PBUF  = alloc(256);
    float* SCAL  = alloc(16);
    (void)ws_size; (void)in_sizes; (void)n_in; (void)out_size;

    const int dimsH[3]  = {H, H, H};
    const int dimsC1[3] = {H, H, C1};

    // ---- level 1 (8192 nodes) ----
    extract_diag<<<dim3((N1 + 255) / 256), dim3(256), 0, stream>>>(adj, N1, DIAGA);
    reduce_partial<<<dim3(256), dim3(256), 0, stream>>>(adj, (size_t)N1 * N1, 1, PBUF);
    reduce_final<<<dim3(1), dim3(256), 0, stream>>>(PBUF, 256, 1.0f, SCAL + 0); // ||adj||_F^2

    run_gnn(stream, d_in, g1p, x, N1, FIN, dimsC1, adj, DIAGA, B1, B2, B3); // s logits -> B3
    run_gnn(stream, d_in, g1e, x, N1, FIN, dimsH,  adj, DIAGA, B1, B2, B2); // h_embed  -> B2

    softmax_ent<<<dim3(N1), dim3(256), 0, stream>>>(B3, C1, ENT);           // s (in place)
    reduce_partial<<<dim3(256), dim3(256), 0, stream>>>(ENT, (size_t)N1, 0, PBUF);
    reduce_final<<<dim3(1), dim3(256), 0, stream>>>(PBUF, 256, 1.0f / (float)N1, SCAL + 6); // ent1

    // diff-pool 1
    launch_gemm(stream, B3, B2, XP,   C1, H,  N1, C1, H,  H,  GF_TRANSA, nullptr, nullptr); // x_p = s^T h
    launch_gemm(stream, adj, B3, B1,  N1, C1, N1, N1, C1, C1, 0, nullptr, nullptr);         // tmp = adj s
    launch_gemm(stream, B3, B1, ADJ1, C1, C1, N1, C1, C1, C1, GF_TRANSA, nullptr, nullptr); // adj1 = s^T tmp
    launch_gemm(stream, B3, B3, G1,   C1, C1, N1, C1, C1, C1, GF_TRANSA, nullptr, nullptr); // G = s^T s
    extract_diag<<<dim3((C1 + 255) / 256), dim3(256), 0, stream>>>(ADJ1, C1, DIAG1);
    trace_kernel<<<dim3(1), dim3(256), 0, stream>>>(ADJ1, C1, SCAL + 1);                    // tr(adj1)
    reduce_partial<<<dim3(256), dim3(256), 0, stream>>>(G1, (size_t)C1 * C1, 1, PBUF);
    reduce_final<<<dim3(1), dim3(256), 0, stream>>>(PBUF, 256, 1.0f, SCAL + 2);             // ||G||^2
    reduce_partial<<<dim3(256), dim3(256), 0, stream>>>(ADJ1, (size_t)C1 * C1, 1, PBUF);
    reduce_final<<<dim3(1), dim3(256), 0, stream>>>(PBUF, 256, 1.0f, SCAL + 3);             // ||adj1||^2

    // ---- level 2 (2048 nodes) ----
    run_gnn(stream, d_in, g2p, XP, C1, H, dimsH, ADJ1, DIAG1, T2, H2, S2);  // s2 logits -> S2
    run_gnn(stream, d_in, g2e, XP, C1, H, dimsH, ADJ1, DIAG1, T2, H2, H2); // h2_embed -> H2

    softmax_ent<<<dim3(C1), dim3(256), 0, stream>>>(S2, C2, ENT);
    reduce_partial<<<dim3(256), dim3(256), 0, stream>>>(ENT, (size_t)C1, 0, PBUF);
    reduce_final<<<dim3(1), dim3(256), 0, stream>>>(PBUF, 256, 1.0f / (float)C1, SCAL + 7); // ent2

    // diff-pool 2
    launch_gemm(stream, S2, H2, XP2,  C2, H,  C1, C2, H,  H,  GF_TRANSA, nullptr, nullptr); // x_p2
    launch_gemm(stream, ADJ1, S2, T2, C1, C2, C1, C1, C2, C2, 0, nullptr, nullptr);         // tmp2 = adj1 s2
    launch_gemm(stream, S2, T2, ADJ2, C2, C2, C1, C2, C2, C2, GF_TRANSA, nullptr, nullptr); // adj2
    launch_gemm(stream, S2, S2, G2,   C2, C2, C1, C2, C2, C2, GF_TRANSA, nullptr, nullptr); // G2
    extract_diag<<<dim3((C2 + 255) / 256), dim3(256), 0, stream>>>(ADJ2, C2, DIAG2);
    trace_kernel<<<dim3(1), dim3(256), 0, stream>>>(ADJ2, C2, SCAL + 4);
    reduce_partial<<<dim3(256), dim3(256), 0, stream>>>(G2, (size_t)C2 * C2, 1, PBUF);
    reduce_final<<<dim3(1), dim3(256), 0, stream>>>(PBUF, 256, 1.0f, SCAL + 5);             // ||G2||^2

    // ---- level 3 (512 nodes) + head ----
    run_gnn(stream, d_in, g3e, XP2, C2, H, dimsH, ADJ2, DIAG2, T2, H2, H2); // h3 -> H2
    head_kernel<<<dim3(1), dim3(512), 0, stream>>>(H2, lin1_w, lin1_b, lin2_w, lin2_b, out);
    finish_kernel<<<dim3(1), dim3(1), 0, stream>>>(SCAL, out);
}